// GinNet_64991445123405
// MI455X (gfx1250) — compile-verified
//
#include <hip/hip_runtime.h>
#include <hip/hip_bf16.h>

typedef __attribute__((ext_vector_type(2))) float v2f;
typedef __attribute__((ext_vector_type(8))) float v8f;

// ---------------------------------------------------------------- utilities
__global__ void zero_kernel(float* __restrict__ p, int n) {
    int i = blockIdx.x * blockDim.x + threadIdx.x;
    if (i < n) p[i] = 0.0f;
}

// identity per-channel scale/shift (for the raw layer-1 input)
__global__ void ident_ss_kernel(float* __restrict__ ss) {
    int c = threadIdx.x;
    ss[c] = 1.0f;
    ss[64 + c] = 0.0f;
}

// ------------------------------------------------- edge gather + scatter-add
// aggr[dst][c] += BN(x[src][c]); feature dim = 1<<dshift (16 or 64).
// Per-channel scale/shift of the producing layer is applied on load (fused BN).
__global__ void gin_scatter_kernel(const float* __restrict__ xin,
                                   const float* __restrict__ ss,
                                   const int* __restrict__ src,
                                   const int* __restrict__ dst,
                                   float* __restrict__ aggr,
                                   int ne, int dshift) {
    int idx = blockIdx.x * blockDim.x + threadIdx.x;
    int total = ne << dshift;
    if (idx >= total) return;
    int e = idx >> dshift;
    int c = idx & ((1 << dshift) - 1);
    int s = src[e];
    int t = dst[e];
    float v = xin[(s << dshift) + c] * ss[c] + ss[64 + c];
    atomicAdd(&aggr[(t << dshift) + c], v);
}

// --------------------------------------------------- fused GIN MLP via WMMA
// out = ReLU( ReLU( ((1+eps)*BN(x) + aggr) @ Wa + ba ) @ Wb + bb )
// KDIM in {16,64}; in-place (out==xin) is safe: block reads its rows to LDS
// before writing.  16 waves: wave w -> rowtile w>>2, coltile w&3.
template <int KDIM>
__global__ __launch_bounds__(512) void gin_mlp_kernel(
    const float* __restrict__ xin, const float* __restrict__ aggr,
    const float* __restrict__ ssin, const float* __restrict__ eps_p,
    const float* __restrict__ Wa, const float* __restrict__ ba,
    const float* __restrict__ Wb, const float* __restrict__ bb,
    float* __restrict__ out, int n) {
    constexpr int KSH = (KDIM == 16) ? 4 : 6;
    __shared__ float sIn[64][68];
    __shared__ float sWa[64][68];
    __shared__ float sWb[64][68];
    __shared__ float sH1[64][68];
    __shared__ float sBa[64];
    __shared__ float sBb[64];

    const int tid  = threadIdx.x;
    const int row0 = blockIdx.x * 64;
    const float ep = 1.0f + eps_p[0];

    for (int i = tid; i < (KDIM << 6); i += 512)
        sWa[i >> 6][i & 63] = Wa[i];
    for (int i = tid; i < 64 * 64; i += 512)
        sWb[i >> 6][i & 63] = Wb[i];
    if (tid < 64) { sBa[tid] = ba[tid]; sBb[tid] = bb[tid]; }

    // input tile, fused (1+eps)*BN(x) + aggr, zero-padded past n
    for (int i = tid; i < (64 << KSH); i += 512) {
        int r = i >> KSH;
        int c = i & (KDIM - 1);
        int gr = row0 + r;
        float v = 0.0f;
        if (gr < n)
            v = ep * (xin[gr * KDIM + c] * ssin[c] + ssin[64 + c]) +
                aggr[gr * KDIM + c];
        sIn[r][c] = v;
    }
    __syncthreads();

    const int wave = tid >> 5;
    const int lane = tid & 31;
    const int half = lane >> 4;  // which 16-lane half
    const int lr   = lane & 15;
    const int rT   = (wave >> 2) << 4;  // 0,16,32,48
    const int cT   = (wave & 3) << 4;   // 0,16,32,48

    // ---- GEMM1: sIn(64 x KDIM) @ sWa(KDIM x 64), fully unrolled
    v8f acc = {};
    #pragma unroll
    for (int k = 0; k < KDIM; k += 4) {
        v2f a, b;
        a.x = sIn[rT + lr][k + 2 * half];
        a.y = sIn[rT + lr][k + 2 * half + 1];
        b.x = sWa[k + 2 * half][cT + lr];
        b.y = sWa[k + 2 * half + 1][cT + lr];
        acc = __builtin_amdgcn_wmma_f32_16x16x4_f32(
            false, a, false, b, (short)0, acc, false, false);
    }
    {
        float bias = sBa[cT + lr];
        #pragma unroll
        for (int r = 0; r < 8; ++r) {
            float v = acc[r] + bias;
            sH1[rT + r + 8 * half][cT + lr] = fmaxf(v, 0.0f);
        }
    }
    __syncthreads();

    // ---- GEMM2: sH1(64 x 64) @ sWb(64 x 64), fully unrolled
    v8f acc2 = {};
    #pragma unroll
    for (int k = 0; k < 64; k += 4) {
        v2f a, b;
        a.x = sH1[rT + lr][k + 2 * half];
        a.y = sH1[rT + lr][k + 2 * half + 1];
        b.x = sWb[k + 2 * half][cT + lr];
        b.y = sWb[k + 2 * half + 1][cT + lr];
        acc2 = __builtin_amdgcn_wmma_f32_16x16x4_f32(
            false, a, false, b, (short)0, acc2, false, false);
    }
    {
        float bias = sBb[cT + lr];
        #pragma unroll
        for (int r = 0; r < 8; ++r) {
            int gr = row0 + rT + r + 8 * half;
            if (gr < n)
                out[gr * 64 + cT + lr] = fmaxf(acc2[r] + bias, 0.0f);
        }
    }
}

// --------------------------------------------------------- batchnorm: stats
// stats[0:64] = sum per channel, stats[64:128] = sum of squares
__global__ void bn_stats_kernel(const float* __restrict__ h,
                                float* __restrict__ stats, int n) {
    int c  = threadIdx.x & 63;
    int rl = threadIdx.x >> 6;  // 0..3
    float s = 0.0f, q = 0.0f;
    for (int r = blockIdx.x * 4 + rl; r < n; r += gridDim.x * 4) {
        float v = h[r * 64 + c];
        s += v;
        q += v * v;
    }
    __shared__ float ls[256];
    __shared__ float lq[256];
    ls[threadIdx.x] = s;
    lq[threadIdx.x] = q;
    __syncthreads();
    if (rl == 0) {
        s = ls[c] + ls[64 + c] + ls[128 + c] + ls[192 + c];
        q = lq[c] + lq[64 + c] + lq[128 + c] + lq[192 + c];
        atomicAdd(&stats[c], s);
        atomicAdd(&stats[64 + c], q);
    }
}

// fold gamma/beta/mean/var -> per-channel scale & shift (applied by consumers)
__global__ void bn_finalize_kernel(const float* __restrict__ stats,
                                   const float* __restrict__ gamma,
                                   const float* __restrict__ beta,
                                   float* __restrict__ ss, float inv_n) {
    int c = threadIdx.x;
    float mu  = stats[c] * inv_n;
    float var = stats[64 + c] * inv_n - mu * mu;
    float s = gamma[c] * rsqrtf(var + 1e-5f);
    ss[c] = s;
    ss[64 + c] = beta[c] - mu * s;
}

// -------------------------------------------------------------- mean pooling
// applies the last layer's BN scale/shift on load (fused)
__global__ void pool_kernel(const float* __restrict__ h,
                            const float* __restrict__ ss,
                            const int* __restrict__ batch,
                            float* __restrict__ pooled,
                            float* __restrict__ cnt, int n) {
    int i = blockIdx.x * blockDim.x + threadIdx.x;
    if (i >= n * 64) return;
    int r = i >> 6;
    int c = i & 63;
    int b = batch[r];
    float v = h[i] * ss[c] + ss[64 + c];
    atomicAdd(&pooled[b * 64 + c], v);
    if (c == 0) atomicAdd(&cnt[b], 1.0f);
}

// ------------------------------------------------------------- readout MLP
__global__ void readout_kernel(const float* __restrict__ pooled,
                               const float* __restrict__ cnt,
                               const float* __restrict__ Wf1,
                               const float* __restrict__ bf1,
                               const float* __restrict__ Wf2,
                               const float* __restrict__ bf2,
                               float* __restrict__ out, int g) {
    int gi = blockIdx.x * blockDim.x + threadIdx.x;
    if (gi >= g) return;
    float inv = 1.0f / fmaxf(cnt[gi], 1.0f);
    float acc[10];
    #pragma unroll
    for (int j = 0; j < 10; ++j) acc[j] = bf1[j];
    for (int c = 0; c < 64; ++c) {
        float v = pooled[gi * 64 + c] * inv;
        #pragma unroll
        for (int j = 0; j < 10; ++j) acc[j] += v * Wf1[c * 10 + j];
    }
    float o = bf2[0];
    #pragma unroll
    for (int j = 0; j < 10; ++j) o += fmaxf(acc[j], 0.0f) * Wf2[j];
    out[gi] = o;
}

// ------------------------------------------------------------- host driver
template <int KDIM>
static void run_gin_layer(const float* xin, const float* ss_in,
                          const int* src, const int* dst,
                          const float* Wa, const float* ba,
                          const float* Wb, const float* bb,
                          const float* eps, const float* gamma,
                          const float* beta,
                          float* h_out, float* aggr, float* stats,
                          float* ss_out, int n, int ne, hipStream_t stream) {
    constexpr int KSH = (KDIM == 16) ? 4 : 6;
    int na = n << KSH;
    zero_kernel<<<(na + 255) / 256, 256, 0, stream>>>(aggr, na);
    int st = ne << KSH;
    gin_scatter_kernel<<<(st + 255) / 256, 256, 0, stream>>>(
        xin, ss_in, src, dst, aggr, ne, KSH);
    gin_mlp_kernel<KDIM><<<(n + 63) / 64, 512, 0, stream>>>(
        xin, aggr, ss_in, eps, Wa, ba, Wb, bb, h_out, n);
    zero_kernel<<<1, 128, 0, stream>>>(stats, 128);
    bn_stats_kernel<<<1024, 256, 0, stream>>>(h_out, stats, n);
    bn_finalize_kernel<<<1, 64, 0, stream>>>(stats, gamma, beta, ss_out,
                                             1.0f / (float)n);
}

extern "C" void kernel_launch(void* const* d_in, const int* in_sizes, int n_in,
                              void* d_out, int out_size, void* d_ws,
                              size_t ws_size, hipStream_t stream) {
    const float* x     = (const float*)d_in[0];
    const int*   ei    = (const int*)d_in[1];
    const int*   batch = (const int*)d_in[2];
    const float* W1a = (const float*)d_in[3];
    const float* b1a = (const float*)d_in[4];
    const float* W1b = (const float*)d_in[5];
    const float* b1b = (const float*)d_in[6];
    const float* e1  = (const float*)d_in[7];
    const float* g1  = (const float*)d_in[8];
    const float* be1 = (const float*)d_in[9];
    const float* W2a = (const float*)d_in[10];
    const float* b2a = (const float*)d_in[11];
    const float* W2b = (const float*)d_in[12];
    const float* b2b = (const float*)d_in[13];
    const float* e2  = (const float*)d_in[14];
    const float* g2  = (const float*)d_in[15];
    const float* be2 = (const float*)d_in[16];
    const float* W3a = (const float*)d_in[17];
    const float* b3a = (const float*)d_in[18];
    const float* W3b = (const float*)d_in[19];
    const float* b3b = (const float*)d_in[20];
    const float* e3  = (const float*)d_in[21];
    const float* g3  = (const float*)d_in[22];
    const float* be3 = (const float*)d_in[23];
    const float* Wf1 = (const float*)d_in[24];
    const float* bf1 = (const float*)d_in[25];
    const float* Wf2 = (const float*)d_in[26];
    const float* bf2 = (const float*)d_in[27];

    const int n = in_sizes[2];        // N nodes (batch array length)
    const int e = in_sizes[1] / 2;    // E edges
    const int g = out_size;           // G graphs (output is [G,1])

    const int* src = ei;
    const int* dst = ei + e;

    float* ws     = (float*)d_ws;
    float* aggr   = ws;                        // N*64
    float* h      = aggr + (size_t)n * 64;     // N*64 (ping, reused in place)
    float* stats  = h + (size_t)n * 64;        // 128
    float* ss     = stats + 128;               // 128 (layer scale/shift)
    float* ss_id  = ss + 128;                  // 128 (identity)
    float* pooled = ss_id + 128;               // G*64
    float* cnt    = pooled + (size_t)g * 64;   // G
    (void)ws_size; (void)n_in;

    ident_ss_kernel<<<1, 64, 0, stream>>>(ss_id);

    // three GIN layers (layers 2 & 3 run in place on h); BN of each layer is
    // folded into per-channel scale/shift consumed by the next stage.
    run_gin_layer<16>(x, ss_id, src, dst, W1a, b1a, W1b, b1b, e1, g1, be1,
                      h, aggr, stats, ss, n, e, stream);
    run_gin_layer<64>(h, ss, src, dst, W2a, b2a, W2b, b2b, e2, g2, be2,
                      h, aggr, stats, ss, n, e, stream);
    run_gin_layer<64>(h, ss, src, dst, W3a, b3a, W3b, b3b, e3, g3, be3,
                      h, aggr, stats, ss, n, e, stream);

    // global mean pool (BN of layer 3 fused on load) + readout
    int pz = g * 64 + g;
    zero_kernel<<<(pz + 255) / 256, 256, 0, stream>>>(pooled, pz);
    int pt = n * 64;
    pool_kernel<<<(pt + 255) / 256, 256, 0, stream>>>(h, ss, batch, pooled,
                                                      cnt, n);
    readout_kernel<<<(g + 63) / 64, 64, 0, stream>>>(
        pooled, cnt, Wf1, bf1, Wf2, bf2, (float*)d_out, g);
}